// ConvAttentionLayer_13331578486816
// MI455X (gfx1250) — compile-verified
//
#include <hip/hip_runtime.h>
#include <hip/hip_bf16.h>
#include <math.h>

#define N_ENT   10000
#define HID     100
#define OC      32
#define E_EDGES 40000
#define NPOS    198            // (HID-1)*2 conv output positions per edge
#define NTILES  13             // ceil(198/16)
#define FC_IN   (OC*NPOS)      // 6336
#define EPS     1e-5f

typedef float v2f __attribute__((ext_vector_type(2)));
typedef float v8f __attribute__((ext_vector_type(8)));

// ---- workspace layout (float offsets) ----
#define WS_BN1   0                      // 2: sum, sumsq of gathered x
#define WS_ZSUM  16                     // 32 per-oc z sums
#define WS_ZSQ   48                     // 32 per-oc z sumsq
#define WS_SCO   80                     // 32 fused scale
#define WS_TCO   112                    // 32 fused shift
#define WS_SCORE 144                    // E
#define WS_M     (WS_SCORE + E_EDGES)   // N_ENT: max(segmax, 1)
#define WS_DEN   (WS_M + N_ENT)         // N_ENT
#define WS_EXPE  (WS_DEN + N_ENT)       // E

__device__ __forceinline__ v8f wmma_k4(v2f a, v2f b, v8f c) {
  // D(16x16) = A(16x4,f32) * B(4x16,f32) + C
  return __builtin_amdgcn_wmma_f32_16x16x4_f32(false, a, false, b, (short)0, c,
                                               false, false);
}

__device__ __forceinline__ void atomicMaxF(float* addr, float val) {
  unsigned int* ua = (unsigned int*)addr;
  unsigned int old = *ua;
  while (__uint_as_float(old) < val) {
    unsigned int assumed = old;
    old = atomicCAS(ua, assumed, __float_as_uint(val));
    if (old == assumed) break;
  }
}

// ---------------- phase 0: zero the small stats region -------------------
__global__ void k_init_stats(float* ws) {
  int i = threadIdx.x;
  if (i < WS_SCORE) ws[i] = 0.0f;
}

// ---------------- phase 1: BN1 global sum / sumsq over gathered x --------
__global__ __launch_bounds__(256) void k_bn1_stats(const int* __restrict__ data,
                                                   const float* __restrict__ ent,
                                                   const float* __restrict__ rel,
                                                   float* ws) {
  float s = 0.f, q = 0.f;
  const int total = E_EDGES * 3 * HID;
  for (int idx = blockIdx.x * blockDim.x + threadIdx.x; idx < total;
       idx += gridDim.x * blockDim.x) {
    int e = idx / (3 * HID);
    int k = idx - e * (3 * HID);
    int col = k / HID, row = k - col * HID;
    int id = data[e * 3 + col];                 // h, r, or t
    float v = (col == 1) ? rel[id * HID + row] : ent[id * HID + row];
    s += v; q += v * v;
  }
  __shared__ float ss[256], sq[256];
  ss[threadIdx.x] = s; sq[threadIdx.x] = q;
  __syncthreads();
  for (int off = 128; off > 0; off >>= 1) {
    if (threadIdx.x < off) {
      ss[threadIdx.x] += ss[threadIdx.x + off];
      sq[threadIdx.x] += sq[threadIdx.x + off];
    }
    __syncthreads();
  }
  if (threadIdx.x == 0) {
    atomicAdd(&ws[WS_BN1 + 0], ss[0]);
    atomicAdd(&ws[WS_BN1 + 1], sq[0]);
  }
}

// ---------------- phase 2: per-oc sum/sumsq of z = conv(raw x) via WMMA --
__global__ __launch_bounds__(256) void k_zstats(const int* __restrict__ data,
                                                const float* __restrict__ ent,
                                                const float* __restrict__ rel,
                                                const float* __restrict__ convw,
                                                float* ws) {
  __shared__ float lds[8 * 312];                // 3 rows x 104 per wave
  const int lane = threadIdx.x & 31;
  float* my = &lds[(threadIdx.x >> 5) * 312];
  const int n = lane & 15;                      // B/D column (oc), A row (pos)
  const int half = lane >> 4;                   // K-pair half per ISA layout
  const int kp = half * 2;

  v2f b0, b1;                                   // conv_w as B (4 x 16), oc 0-15 / 16-31
  b0.x = convw[n * 4 + kp];        b0.y = convw[n * 4 + kp + 1];
  b1.x = convw[(n + 16) * 4 + kp]; b1.y = convw[(n + 16) * 4 + kp + 1];

  float sum0 = 0.f, sq0 = 0.f, sum1 = 0.f, sq1 = 0.f;
  const int gw = (blockIdx.x * blockDim.x + threadIdx.x) >> 5;
  const int nw = (gridDim.x * blockDim.x) >> 5;

  for (int e = gw; e < E_EDGES; e += nw) {
    int h = data[e * 3 + 0], r = data[e * 3 + 1], t = data[e * 3 + 2];
    for (int k = lane; k < HID; k += 32) {
      my[k]       = ent[h * HID + k];
      my[104 + k] = rel[r * HID + k];
      my[208 + k] = ent[t * HID + k];
    }
    asm volatile("s_wait_dscnt 0x0" ::: "memory");

    for (int tile = 0; tile < NTILES; ++tile) {
      int pos = tile * 16 + n;
      v2f a; a.x = 0.f; a.y = 0.f;
      if (pos < NPOS) {
        int i = pos >> 1, j = pos & 1;
        int row = i + half;                     // ki = half
        a.x = my[j * 104 + row];                // tap (ki, 0)
        a.y = my[(j + 1) * 104 + row];          // tap (ki, 1)
      }
      v8f c0 = {}; v8f c1 = {};
      c0 = wmma_k4(a, b0, c0);
      c1 = wmma_k4(a, b1, c1);
#pragma unroll
      for (int rr = 0; rr < 8; ++rr) {
        int posr = tile * 16 + rr + half * 8;
        if (posr < NPOS) {
          float z0 = c0[rr], z1 = c1[rr];
          sum0 += z0; sq0 += z0 * z0;
          sum1 += z1; sq1 += z1 * z1;
        }
      }
    }
  }
  // lanes L and L+16 hold the same oc
  sum0 += __shfl_xor(sum0, 16, 32);
  sq0  += __shfl_xor(sq0, 16, 32);
  sum1 += __shfl_xor(sum1, 16, 32);
  sq1  += __shfl_xor(sq1, 16, 32);
  if (lane < 16) {
    atomicAdd(&ws[WS_ZSUM + lane], sum0);
    atomicAdd(&ws[WS_ZSQ + lane], sq0);
    atomicAdd(&ws[WS_ZSUM + 16 + lane], sum1);
    atomicAdd(&ws[WS_ZSQ + 16 + lane], sq1);
  }
}

// ---------------- phase 3: fold BN1+BN2 into per-oc affine (s, t) --------
__global__ void k_coeffs(float* ws, const float* bn1g, const float* bn2g,
                         const float* bn2b) {
  int oc = threadIdx.x;
  if (oc >= OC) return;
  const float cnt1 = (float)(E_EDGES * 3 * HID);
  float m1 = ws[WS_BN1] / cnt1;
  float v1 = ws[WS_BN1 + 1] / cnt1 - m1 * m1;
  float a = bn1g[0] * rsqrtf(v1 + EPS);         // BN1 scale (shift cancels)
  const float cntz = (float)(E_EDGES * NPOS);
  float mz = ws[WS_ZSUM + oc] / cntz;
  float vz = ws[WS_ZSQ + oc] / cntz - mz * mz;
  float r2 = rsqrtf(a * a * vz + EPS);
  float s = a * bn2g[oc] * r2;
  float t = bn2b[oc] - s * mz;                  // conv_b / BN1-beta cancel in BN2
  ws[WS_SCO + oc] = s;
  ws[WS_TCO + oc] = t;
}

// ---------------- phase 4: fused conv+BN+ReLU+fc score via WMMA ----------
__global__ __launch_bounds__(256) void k_score(const int* __restrict__ data,
                                               const float* __restrict__ ent,
                                               const float* __restrict__ rel,
                                               const float* __restrict__ convw,
                                               const float* __restrict__ fcw,
                                               float* ws) {
  __shared__ float fw[FC_IN];                   // 25.3 KB of 320 KB LDS
  __shared__ float lds[8 * 312];
  for (int idx = threadIdx.x; idx < FC_IN; idx += blockDim.x) fw[idx] = fcw[idx];
  __syncthreads();

  const int lane = threadIdx.x & 31;
  float* my = &lds[(threadIdx.x >> 5) * 312];
  const int n = lane & 15, half = lane >> 4, kp = half * 2;

  v2f b0, b1;
  b0.x = convw[n * 4 + kp];        b0.y = convw[n * 4 + kp + 1];
  b1.x = convw[(n + 16) * 4 + kp]; b1.y = convw[(n + 16) * 4 + kp + 1];
  const float s0 = ws[WS_SCO + n],      t0 = ws[WS_TCO + n];
  const float s1 = ws[WS_SCO + 16 + n], t1 = ws[WS_TCO + 16 + n];

  const int gw = (blockIdx.x * blockDim.x + threadIdx.x) >> 5;
  const int nw = (gridDim.x * blockDim.x) >> 5;

  for (int e = gw; e < E_EDGES; e += nw) {
    int h = data[e * 3 + 0], r = data[e * 3 + 1], t = data[e * 3 + 2];
    for (int k = lane; k < HID; k += 32) {
      my[k]       = ent[h * HID + k];
      my[104 + k] = rel[r * HID + k];
      my[208 + k] = ent[t * HID + k];
    }
    asm volatile("s_wait_dscnt 0x0" ::: "memory");

    float sc = 0.f;
    for (int tile = 0; tile < NTILES; ++tile) {
      int pos = tile * 16 + n;
      v2f a; a.x = 0.f; a.y = 0.f;
      if (pos < NPOS) {
        int i = pos >> 1, j = pos & 1;
        int row = i + half;
        a.x = my[j * 104 + row];
        a.y = my[(j + 1) * 104 + row];
      }
      v8f c0 = {}; v8f c1 = {};
      c0 = wmma_k4(a, b0, c0);
      c1 = wmma_k4(a, b1, c1);
#pragma unroll
      for (int rr = 0; rr < 8; ++rr) {
        int posr = tile * 16 + rr + half * 8;
        if (posr < NPOS) {
          float u0 = s0 * c0[rr] + t0;
          if (u0 > 0.f) sc += fw[n * NPOS + posr] * u0;
          float u1 = s1 * c1[rr] + t1;
          if (u1 > 0.f) sc += fw[(n + 16) * NPOS + posr] * u1;
        }
      }
    }
    for (int off = 16; off > 0; off >>= 1) sc += __shfl_xor(sc, off, 32);
    if (lane == 0) ws[WS_SCORE + e] = sc;
  }
}

// ---------------- phase 5: segment softmax + attention scatter -----------
__global__ void k_init_m(float* ws) {
  int v = blockIdx.x * blockDim.x + threadIdx.x;
  if (v < N_ENT) ws[WS_M + v] = 1.0f;           // max(segmax, 1); empty seg -> 1
}

__global__ void k_segmax(const int* __restrict__ data, float* ws) {
  int e = blockIdx.x * blockDim.x + threadIdx.x;
  if (e < E_EDGES) atomicMaxF(&ws[WS_M + data[e * 3]], ws[WS_SCORE + e]);
}

__global__ void k_den_init(float* ws) {
  int v = blockIdx.x * blockDim.x + threadIdx.x;
  if (v < N_ENT) ws[WS_DEN + v] = __expf(1.0f - ws[WS_M + v]);
}

__global__ void k_expe(const int* __restrict__ data, float* ws) {
  int e = blockIdx.x * blockDim.x + threadIdx.x;
  if (e < E_EDGES) {
    int h = data[e * 3];
    float ee = __expf(ws[WS_SCORE + e] - ws[WS_M + h]);
    ws[WS_EXPE + e] = ee;
    atomicAdd(&ws[WS_DEN + h], ee);
  }
}

__global__ void k_outself(const float* __restrict__ ws,
                          const float* __restrict__ ent, float* out) {
  int idx = blockIdx.x * blockDim.x + threadIdx.x;
  if (idx < N_ENT * HID) {
    int v = idx / HID;
    float att = __expf(1.0f - ws[WS_M + v]) / ws[WS_DEN + v];
    out[idx] = att * ent[idx];
  }
}

__global__ void k_scatter(const int* __restrict__ data,
                          const float* __restrict__ ws,
                          const float* __restrict__ ent, float* out) {
  int idx = blockIdx.x * blockDim.x + threadIdx.x;
  if (idx < E_EDGES * HID) {
    int e = idx / HID, d = idx - e * HID;
    int h = data[e * 3], t = data[e * 3 + 2];
    float att = ws[WS_EXPE + e] / ws[WS_DEN + h];
    atomicAdd(&out[h * HID + d], att * ent[t * HID + d]);
  }
}

extern "C" void kernel_launch(void* const* d_in, const int* in_sizes, int n_in,
                              void* d_out, int out_size, void* d_ws, size_t ws_size,
                              hipStream_t stream) {
  const int*   data  = (const int*)d_in[0];
  const float* ent   = (const float*)d_in[1];
  const float* rel   = (const float*)d_in[2];
  const float* convw = (const float*)d_in[3];
  // d_in[4] = conv_b, d_in[6] = bn1_beta: provably cancel inside BN2
  const float* bn1g  = (const float*)d_in[5];
  const float* bn2g  = (const float*)d_in[7];
  const float* bn2b  = (const float*)d_in[8];
  const float* fcw   = (const float*)d_in[9];
  float* ws  = (float*)d_ws;
  float* out = (float*)d_out;
  (void)in_sizes; (void)n_in; (void)out_size; (void)ws_size;

  k_init_stats<<<1, 256, 0, stream>>>(ws);
  k_bn1_stats<<<256, 256, 0, stream>>>(data, ent, rel, ws);
  k_zstats<<<256, 256, 0, stream>>>(data, ent, rel, convw, ws);
  k_coeffs<<<1, 64, 0, stream>>>(ws, bn1g, bn2g, bn2b);
  k_score<<<256, 256, 0, stream>>>(data, ent, rel, convw, fcw, ws);
  k_init_m<<<(N_ENT + 255) / 256, 256, 0, stream>>>(ws);
  k_segmax<<<(E_EDGES + 255) / 256, 256, 0, stream>>>(data, ws);
  k_den_init<<<(N_ENT + 255) / 256, 256, 0, stream>>>(ws);
  k_expe<<<(E_EDGES + 255) / 256, 256, 0, stream>>>(data, ws);
  k_outself<<<(N_ENT * HID + 255) / 256, 256, 0, stream>>>(ws, ent, out);
  k_scatter<<<(E_EDGES * HID + 255) / 256, 256, 0, stream>>>(data, ws, ent, out);
}